// T5RelativePositionBias_50285477102159
// MI455X (gfx1250) — compile-verified
//
#include <hip/hip_runtime.h>
#include <math.h>

// ---------------------------------------------------------------------------
// T5 relative position bias, MI455X (gfx1250).
//
// out[q][k] = table[bucket(k-q)] * 0.125  over 8192x8192  (Toeplitz in d=k-q)
//
// Roofline: 256 MiB fp32 stores / 23.3 TB/s  ->  ~11.5 us floor; compute ~0.
// Plan:
//   K1: build 16383-entry diagonal LUT via exact f32 WMMA one-hot @ table
//       (V_WMMA_F32_16X16X4_F32) -- mirrors the original T5 matmul, bit-exact.
//   K2: stream 128-bit non-temporal stores, reading the 64KB LUT (cache-hot).
// ---------------------------------------------------------------------------

typedef __attribute__((ext_vector_type(2))) float v2f;
typedef __attribute__((ext_vector_type(4))) float f4;
typedef __attribute__((ext_vector_type(8))) float v8f;

#define NUM_BUCKETS 32
#define HALF_BUCKETS 16   // non-causal: num_buckets // 2
#define MAX_EXACT 8       // HALF_BUCKETS // 2
#define SCALE 0.125f
// log(MAX_DISTANCE / max_exact) = log(128/8) = log(16)
#define LOG_RATIO 2.7725887222397812f

// Faithful fp32 port of _relative_position_bucket (non-causal, int32 trunc).
__device__ __forceinline__ int t5_bucket(int rel /* k - q */) {
  int n = -rel;                       // n = -relative_position
  int ret = (n < 0) ? HALF_BUCKETS : 0;
  n = (n < 0) ? -n : n;
  int b;
  if (n < MAX_EXACT) {
    b = n;
  } else {
    float nf = (float)n;
    int v = MAX_EXACT +
            (int)(logf(nf / (float)MAX_EXACT) / LOG_RATIO * (float)(HALF_BUCKETS - MAX_EXACT));
    b = (v < HALF_BUCKETS - 1) ? v : (HALF_BUCKETS - 1);
  }
  return ret + b;
}

// ---------------------------------------------------------------------------
// Kernel 1: LUT over diagonals via one-hot @ table with f32 WMMA.
//   One wave (32 lanes) produces 16 LUT entries (8 WMMAs, K=32 buckets).
//   A (16x4 f32): A[i][kk] = table[kk]  (same for every row i)
//       ISA layout: lanes 0-15 hold K=0(V0),K=1(V1); lanes 16-31 K=2,K=3.
//   B (4x16 f32): B[kk][j] = (bucket(d0+j)==kk) ? 1 : 0
//       mirrored layout: V_r holds K = r + 2*(lane/16), N = lane%16.
//   D (16x16): every row == LUT chunk; lane (N=lane%16) takes c[0].
// Exact: each dot product has exactly one nonzero fp32 term.
// ---------------------------------------------------------------------------
__global__ void t5_lut_wmma(const float* __restrict__ table,
                            float* __restrict__ lut, int lut_n, int d_off) {
  const int lane = threadIdx.x;          // 0..31, blockDim.x == 32
  const int j = lane & 15;               // output column / LUT sub-index
  const int khalf = (lane >> 4) << 1;    // 0 for lanes 0-15, 2 for lanes 16-31
  const int dj = blockIdx.x * 16 + j;    // LUT index
  const int bucket = t5_bucket(dj - d_off);

  v8f c = {};
#pragma unroll
  for (int base = 0; base < NUM_BUCKETS; base += 4) {
    v2f a, b;
    a.x = table[base + khalf + 0];                       // A: broadcast table
    a.y = table[base + khalf + 1];
    b.x = (bucket == base + khalf + 0) ? 1.0f : 0.0f;    // B: one-hot column
    b.y = (bucket == base + khalf + 1) ? 1.0f : 0.0f;
    c = __builtin_amdgcn_wmma_f32_16x16x4_f32(
        /*neg_a=*/false, a, /*neg_b=*/false, b,
        /*c_mod=*/(short)0, c, /*reuse_a=*/false, /*reuse_b=*/false);
  }
  if (lane < 16 && dj < lut_n) lut[dj] = c[0] * SCALE;
}

// ---------------------------------------------------------------------------
// Kernel 2: store-bandwidth-bound streamer.
//   out[q][k..k+3] = lut[k - q + d_off ..]; 16B NT stores (output > L2).
//   LUT window (64KB) stays hot in WGP$/L2; 4-byte-aligned gather loads.
// ---------------------------------------------------------------------------
__global__ void t5_stream4(const float* __restrict__ lut,
                           float* __restrict__ out, int k_len, int d_off) {
  const int q = blockIdx.y;
  const int k = (blockIdx.x * blockDim.x + threadIdx.x) << 2;
  if (k >= k_len) return;
  const float* __restrict__ src = lut + (k - q + d_off);
  f4 v;
  v.x = src[0];
  v.y = src[1];
  v.z = src[2];
  v.w = src[3];
  f4* dst = (f4*)(out + (size_t)q * (size_t)k_len + (size_t)k);
  __builtin_nontemporal_store(v, dst);   // global_store_b128 ... th:TH_STORE_NT
}

// ---------------------------------------------------------------------------
// Fallback: direct per-element compute (no workspace / odd k_len).
// ---------------------------------------------------------------------------
__global__ void t5_direct(const float* __restrict__ table,
                          float* __restrict__ out, int q_len, int k_len) {
  size_t tid = (size_t)blockIdx.x * blockDim.x + threadIdx.x;
  size_t total = (size_t)q_len * (size_t)k_len;
  if (tid >= total) return;
  int q = (int)(tid / (size_t)k_len);
  int k = (int)(tid - (size_t)q * (size_t)k_len);
  float v = table[t5_bucket(k - q)] * SCALE;
  __builtin_nontemporal_store(v, out + tid);
}

extern "C" void kernel_launch(void* const* d_in, const int* in_sizes, int n_in,
                              void* d_out, int out_size, void* d_ws, size_t ws_size,
                              hipStream_t stream) {
  // inputs: 0 = q_pos (arange), 1 = k_pos (arange), 2 = rel_bias_table (f32 [32,1])
  const float* table = (const float*)d_in[2];
  float* out = (float*)d_out;
  const int q_len = in_sizes[0];
  const int k_len = in_sizes[1];
  const int lut_n = q_len + k_len - 1;   // diagonals d = k - q + (q_len-1)
  const int d_off = q_len - 1;
  const size_t lut_bytes = (size_t)lut_n * sizeof(float);

  if (ws_size >= lut_bytes && (k_len & 3) == 0 && q_len <= 65535) {
    float* lut = (float*)d_ws;
    // K1: one wave per 16 diagonals, exact one-hot @ table via f32 WMMA.
    const int tiles = (lut_n + 15) / 16;
    t5_lut_wmma<<<tiles, 32, 0, stream>>>(table, lut, lut_n, d_off);
    // K2: 256 threads/block along k (each writes a float4); one row per blockIdx.y.
    const int k4_per_row = k_len >> 2;                 // 2048 for 8192
    const int bx = (k4_per_row + 255) / 256;           // 8
    dim3 grid(bx, q_len, 1);
    t5_stream4<<<grid, 256, 0, stream>>>(lut, out, k_len, d_off);
  } else {
    size_t total = (size_t)q_len * (size_t)k_len;
    size_t blocks = (total + 255) / 256;
    t5_direct<<<(unsigned)blocks, 256, 0, stream>>>(table, out, q_len, k_len);
  }
}